// UpdateBlockTDU_OV_81441169866994
// MI455X (gfx1250) — compile-verified
//
#include <hip/hip_runtime.h>
#include <math.h>

// ---------------------------------------------------------------------------
// MI455X (gfx1250) implementation. fp32 WMMA (V_WMMA_F32_16X16X4_F32) for all
// GEMM-shaped work; device-side run-length segmentation; single-workgroup GRU
// scans; dilated TCN as shifted accumulating GEMM passes.
// Round 3: K-loop peeled + pure pointer-increment addressing so the compiler
// strength-reduces all fragment addresses (round 2 recomputed 64-bit addresses
// from the k index every iteration).
// ---------------------------------------------------------------------------

typedef __attribute__((ext_vector_type(2))) float v2f;
typedef __attribute__((ext_vector_type(8))) float v8f;

#define T_SZ   8192
#define D_SZ   512
#define M_ACT  64
#define C_CLS  32
#define HH_SZ  256
#define L_TCN  6
#define SMAX   4096

#define GF_RELU 1
#define GF_ACC  2

// ------------------------------ generic WMMA GEMM --------------------------
// C[m][n] = sum_k A[m+aShift][k] * B[bOff + k*ldbK + n*ldbN]   (+bias/res/acc)
// Requirements (all call sites satisfy these):
//  * K multiple of 4, OR (dyn-K) A columns [K, round4(K)) are zero and B rows
//    [K, round4(K)) are readable finite memory.
//  * A rows outside [0, Mv) and B cols outside [0, N) masked via clamped
//    pointer + 0/1 multiply (branch-free); surrounding memory readable.
__global__ __launch_bounds__(256) void gemm_wmma(
    const float* __restrict__ A, int lda, int aShift,
    const float* __restrict__ B, int bOff, int ldbK, int ldbN,
    const float* __restrict__ bias,
    const float* __restrict__ res, int ldr,
    float* __restrict__ Cm, int ldc,
    int Mfix, const int* __restrict__ Mdyn,
    int N,
    int Kfix, const int* __restrict__ Kdyn,
    int flags)
{
  int Mv = Mdyn ? *Mdyn : Mfix;
  int m0b = (int)blockIdx.y * 64;
  if (m0b >= Mv) return;
  int Kv = Kdyn ? *Kdyn : Kfix;

  int lane = threadIdx.x & 31;
  int wv   = threadIdx.x >> 5;
  int wm   = wv >> 2;            // 0..1   (waves: 2 in M x 4 in N)
  int wn   = wv & 3;             // 0..3
  int half = lane >> 4;          // 0/1
  int l15  = lane & 15;
  int mBase = m0b + wm * 32;
  int nBase = (int)blockIdx.x * 128 + wn * 32;
  int kOff  = half << 1;         // this half's K-pair offset within a 4-step

  // Per-tile clamped base pointers + masks (computed once, branch-free loop).
  float aMask0, aMask1, bMask0, bMask1;
  const float *a0, *a1, *b0l, *b0h, *b1l, *b1h;
  {
    int r0 = mBase + l15 + aShift;
    int r1 = r0 + 16;
    aMask0 = (r0 >= 0 && r0 < Mv) ? 1.0f : 0.0f;
    aMask1 = (r1 >= 0 && r1 < Mv) ? 1.0f : 0.0f;
    int rc0 = r0 < 0 ? 0 : (r0 >= Mv ? Mv - 1 : r0);
    int rc1 = r1 < 0 ? 0 : (r1 >= Mv ? Mv - 1 : r1);
    a0 = A + (size_t)rc0 * lda + kOff;
    a1 = A + (size_t)rc1 * lda + kOff;
    int c0 = nBase + l15;
    int c1 = c0 + 16;
    bMask0 = (c0 < N) ? 1.0f : 0.0f;
    bMask1 = (c1 < N) ? 1.0f : 0.0f;
    int cc0 = c0 < N ? c0 : N - 1;
    int cc1 = c1 < N ? c1 : N - 1;
    const float* bb = B + (size_t)bOff + (size_t)kOff * ldbK;
    b0l = bb + (size_t)cc0 * ldbN;  b0h = b0l + ldbK;
    b1l = bb + (size_t)cc1 * ldbN;  b1h = b1l + ldbK;
  }
  const int bInc = 4 * ldbK;

  v8f acc00 = {}, acc01 = {}, acc10 = {}, acc11 = {};
  int kt = (Kv + 3) >> 2;

  // prologue: load first fragments
  v2f aC0 = *(const v2f*)a0;  aC0.x *= aMask0; aC0.y *= aMask0;
  v2f aC1 = *(const v2f*)a1;  aC1.x *= aMask1; aC1.y *= aMask1;
  v2f bC0; bC0.x = b0l[0] * bMask0; bC0.y = b0h[0] * bMask0;
  v2f bC1; bC1.x = b1l[0] * bMask1; bC1.y = b1h[0] * bMask1;

  for (int kk = 0; kk + 1 < kt; ++kk) {
    // advance and prefetch next step's fragments (pure pointer increments)
    a0 += 4; a1 += 4; b0l += bInc; b0h += bInc; b1l += bInc; b1h += bInc;
    v2f aN0 = *(const v2f*)a0;  aN0.x *= aMask0; aN0.y *= aMask0;
    v2f aN1 = *(const v2f*)a1;  aN1.x *= aMask1; aN1.y *= aMask1;
    v2f bN0; bN0.x = b0l[0] * bMask0; bN0.y = b0h[0] * bMask0;
    v2f bN1; bN1.x = b1l[0] * bMask1; bN1.y = b1h[0] * bMask1;

    acc00 = __builtin_amdgcn_wmma_f32_16x16x4_f32(false, aC0, false, bC0, (short)0, acc00, false, false);
    acc01 = __builtin_amdgcn_wmma_f32_16x16x4_f32(false, aC0, false, bC1, (short)0, acc01, false, false);
    acc10 = __builtin_amdgcn_wmma_f32_16x16x4_f32(false, aC1, false, bC0, (short)0, acc10, false, false);
    acc11 = __builtin_amdgcn_wmma_f32_16x16x4_f32(false, aC1, false, bC1, (short)0, acc11, false, false);

    aC0 = aN0; aC1 = aN1; bC0 = bN0; bC1 = bN1;
  }
  // peeled last step (no prefetch -> no overread, no selects)
  acc00 = __builtin_amdgcn_wmma_f32_16x16x4_f32(false, aC0, false, bC0, (short)0, acc00, false, false);
  acc01 = __builtin_amdgcn_wmma_f32_16x16x4_f32(false, aC0, false, bC1, (short)0, acc01, false, false);
  acc10 = __builtin_amdgcn_wmma_f32_16x16x4_f32(false, aC1, false, bC0, (short)0, acc10, false, false);
  acc11 = __builtin_amdgcn_wmma_f32_16x16x4_f32(false, aC1, false, bC1, (short)0, acc11, false, false);

  // epilogue: C layout per ISA: VGPR v -> row (mBase+mi*16 + v + 8*half), col l15
  v8f accs[2][2] = {{acc00, acc01}, {acc10, acc11}};
#pragma unroll
  for (int mi = 0; mi < 2; ++mi) {
#pragma unroll
    for (int ni = 0; ni < 2; ++ni) {
      int cc = nBase + ni * 16 + l15;
      if (cc >= N) continue;
      float bv = bias ? bias[cc] : 0.0f;
#pragma unroll
      for (int v = 0; v < 8; ++v) {
        int rr = mBase + mi * 16 + v + half * 8;
        size_t ci = (size_t)rr * ldc + cc;
        float val = accs[mi][ni][v] + bv;
        if (flags & GF_ACC) val += Cm[ci];
        if (res) val += res[(size_t)rr * ldr + cc];
        if (flags & GF_RELU) val = fmaxf(val, 0.0f);
        Cm[ci] = val;
      }
    }
  }
}

// ------------------------------ text row normalize -------------------------
__global__ __launch_bounds__(256) void norm_text(const float* __restrict__ t,
                                                 float* __restrict__ o) {
  int c = blockIdx.x, tid = threadIdx.x;
  __shared__ float red[256];
  const float* tr = t + (size_t)c * 512;
  float a = tr[tid], b = tr[tid + 256];
  red[tid] = a * a + b * b;
  __syncthreads();
  for (int off = 128; off > 0; off >>= 1) {
    if (tid < off) red[tid] += red[tid + off];
    __syncthreads();
  }
  float inv = 1.0f / sqrtf(red[0]);
  o[(size_t)c * 512 + tid] = a * inv;
  o[(size_t)c * 512 + tid + 256] = b * inv;
}

// ------------------- row-normalize + dot with text (+argmax) ---------------
__global__ __launch_bounds__(256) void sim_rows(const float* __restrict__ V,
                                                const float* __restrict__ Tn,
                                                float* __restrict__ out,
                                                int rowsFix, const int* rowsDyn,
                                                int* __restrict__ amax) {
  int rows = rowsDyn ? *rowsDyn : rowsFix;
  int r = blockIdx.x;
  if (r >= rows) return;
  __shared__ float v[512];
  __shared__ float red[256];
  __shared__ float sims[32];
  int tid = threadIdx.x;
  const float* vr = V + (size_t)r * 512;
  v[tid] = vr[tid];
  v[tid + 256] = vr[tid + 256];
  __syncthreads();
  red[tid] = v[tid] * v[tid] + v[tid + 256] * v[tid + 256];
  __syncthreads();
  for (int off = 128; off > 0; off >>= 1) {
    if (tid < off) red[tid] += red[tid + off];
    __syncthreads();
  }
  float inv = 1.0f / sqrtf(red[0]);
  __syncthreads();
  int g = tid >> 3, l = tid & 7;   // 32 groups x 8 lanes, one class each
  const float* tg = Tn + (size_t)g * 512;
  float dot = 0.0f;
  for (int h = l; h < 512; h += 8) dot += v[h] * tg[h];
  red[tid] = dot;
  __syncthreads();
  if (l == 0) {
    float s = 0.0f;
#pragma unroll
    for (int q = 0; q < 8; ++q) s += red[g * 8 + q];
    sims[g] = s * inv;   // text already normalized; TEMP = 1
  }
  __syncthreads();
  if (out && tid < 32) out[(size_t)r * 32 + tid] = sims[tid];
  if (amax && tid == 0) {
    int best = 0; float bv = sims[0];
    for (int c = 1; c < 32; ++c) if (sims[c] > bv) { bv = sims[c]; best = c; }
    amax[r] = best;
  }
}

// --------------- run-length segmentation (single workgroup scan) -----------
__global__ __launch_bounds__(1024) void segment_scan(const int* __restrict__ pred,
                                                     int* __restrict__ segid,
                                                     int* __restrict__ starts,
                                                     int* __restrict__ centers,
                                                     int* __restrict__ Sdev) {
  __shared__ int sd[1024];
  int tid = threadIdx.x;
  int carry = 0;
  for (int base = 0; base < T_SZ; base += 1024) {
    int t = base + tid;
    int p = pred[t];
    int chg = (t > 0 && p != pred[t - 1]) ? 1 : 0;
    sd[tid] = chg;
    __syncthreads();
    for (int off = 1; off < 1024; off <<= 1) {
      int v = sd[tid];
      int add = (tid >= off) ? sd[tid - off] : 0;
      __syncthreads();
      sd[tid] = v + add;
      __syncthreads();
    }
    int sid = carry + sd[tid];
    segid[t] = (sid < SMAX) ? sid : (SMAX - 1);
    if ((t == 0 || chg) && sid < SMAX) starts[sid] = t;
    int tot = sd[1023];
    __syncthreads();
    carry += tot;
  }
  __threadfence();
  __syncthreads();
  int S = carry + 1;
  if (S > SMAX) S = SMAX;
  if (tid == 0) Sdev[0] = S;
  for (int s = tid; s < S; s += 1024) {
    int st = starts[s];
    int en = (s + 1 < S) ? (starts[s + 1] - 1) : (T_SZ - 1);
    centers[s] = (st + en) >> 1;
  }
}

// ------------------------------ segment mean pool --------------------------
__global__ __launch_bounds__(256) void seg_mean(const float* __restrict__ ff,
                                                const int* __restrict__ starts,
                                                const int* __restrict__ Sdev,
                                                float* __restrict__ seg) {
  int s = blockIdx.x, S = *Sdev;
  if (s >= S) return;
  int st = starts[s];
  int en = (s + 1 < S) ? (starts[s + 1] - 1) : (T_SZ - 1);
  float inv = 1.0f / (float)(en - st + 1);
  for (int d = threadIdx.x; d < 512; d += 256) {
    float a = 0.0f;
    for (int t = st; t <= en; ++t) a += ff[(size_t)t * 512 + d];
    seg[(size_t)s * 512 + d] = a * inv;
  }
}

__global__ __launch_bounds__(256) void gather_pos(const float* __restrict__ fp,
                                                  const int* __restrict__ centers,
                                                  const int* __restrict__ Sdev,
                                                  float* __restrict__ sp) {
  int s = blockIdx.x;
  if (s >= *Sdev) return;
  int c = centers[s];
  for (int d = threadIdx.x; d < 512; d += 256)
    sp[(size_t)s * 512 + d] = fp[(size_t)c * 512 + d];
}

// ------------------------------ GRU scan (B = 1) ---------------------------
// blockDim == H; each thread computes 3 gate dots (columns j, H+j, 2H+j).
__global__ void gru_scan(const float* __restrict__ gi,
                         const float* __restrict__ Whh,
                         const float* __restrict__ bhh,
                         float* __restrict__ out,
                         int H, int stepsFix, const int* stepsDyn, int reverse) {
  extern __shared__ float h[];
  int S = stepsDyn ? *stepsDyn : stepsFix;
  int j = threadIdx.x;
  int G = 3 * H;
  h[j] = 0.0f;
  __syncthreads();
  for (int s = 0; s < S; ++s) {
    int t = reverse ? (S - 1 - s) : s;
    float hr = bhh[j], hz = bhh[H + j], hn = bhh[2 * H + j];
    for (int k = 0; k < H; ++k) {
      float hk = h[k];
      const float* w = Whh + (size_t)k * G;
      hr += hk * w[j];
      hz += hk * w[H + j];
      hn += hk * w[2 * H + j];
    }
    const float* g = gi + (size_t)t * G;
    float r = 1.0f / (1.0f + expf(-(g[j] + hr)));
    float z = 1.0f / (1.0f + expf(-(g[H + j] + hz)));
    float n = tanhf(g[2 * H + j] + r * hn);
    float hold = h[j];
    float hnew = (1.0f - z) * n + z * hold;
    __syncthreads();
    h[j] = hnew;
    out[(size_t)t * H + j] = hnew;
    __syncthreads();
  }
}

// ------------------------------ small elementwise --------------------------
__global__ __launch_bounds__(256) void segcat_relu(const float* __restrict__ hf,
                                                   const float* __restrict__ hb,
                                                   float* __restrict__ o,
                                                   const int* __restrict__ Sdev) {
  int i = blockIdx.x * 256 + threadIdx.x;
  int t = i >> 9, d = i & 511;
  if (t >= *Sdev) return;
  float v = (d < 256) ? hf[(size_t)t * 256 + d] : hb[(size_t)t * 256 + (d - 256)];
  o[i] = fmaxf(v, 0.0f);
}

__global__ __launch_bounds__(256) void add_rows(const float* __restrict__ a,
                                                const float* __restrict__ b,
                                                float* __restrict__ o,
                                                int rowsFix, const int* rowsDyn) {
  int rows = rowsDyn ? *rowsDyn : rowsFix;
  int i = blockIdx.x * 256 + threadIdx.x;
  if (i >= rows * 512) return;
  o[i] = a[i] + b[i];
}

__global__ __launch_bounds__(256) void upsample_cat(const float* __restrict__ seg2,
                                                    const float* __restrict__ ff,
                                                    const int* __restrict__ segid,
                                                    float* __restrict__ cat2) {
  int i = blockIdx.x * 256 + threadIdx.x;  // over T*512
  int t = i >> 9, d = i & 511;
  cat2[(size_t)t * 1024 + d] = seg2[(size_t)segid[t] * 512 + d];
  cat2[(size_t)t * 1024 + 512 + d] = ff[i];
}

// ------------------------------ row softmax (in place) ---------------------
__global__ __launch_bounds__(256) void softmax_rows(float* __restrict__ mat, int ld,
                                                    int rowsFix, const int* rowsDyn,
                                                    int validFix, const int* validDyn,
                                                    float scale) {
  int rows = rowsDyn ? *rowsDyn : rowsFix;
  int r = blockIdx.x;
  if (r >= rows) return;
  int valid = validDyn ? *validDyn : validFix;
  float* row = mat + (size_t)r * ld;
  __shared__ float red[256];
  int tid = threadIdx.x;
  float m = -3.4e38f;
  for (int i = tid; i < valid; i += 256) m = fmaxf(m, row[i] * scale);
  red[tid] = m;
  __syncthreads();
  for (int o = 128; o > 0; o >>= 1) { if (tid < o) red[tid] = fmaxf(red[tid], red[tid + o]); __syncthreads(); }
  float M = red[0];
  __syncthreads();
  float s = 0.0f;
  for (int i = tid; i < valid; i += 256) s += expf(row[i] * scale - M);
  red[tid] = s;
  __syncthreads();
  for (int o = 128; o > 0; o >>= 1) { if (tid < o) red[tid] += red[tid + o]; __syncthreads(); }
  float inv = 1.0f / red[0];
  for (int i = tid; i < ld; i += 256)
    row[i] = (i < valid) ? expf(row[i] * scale - M) * inv : 0.0f;  // zero-pad for dyn-K GEMM
}

// ===========================================================================
static inline void gemm(hipStream_t s, const float* A, int lda, int aShift,
                        const float* B, int bOff, int ldbK, int ldbN,
                        const float* bias, const float* res, int ldr,
                        float* C, int ldc, int Mcap, int Mfix, const int* Mdyn,
                        int N, int Kfix, const int* Kdyn, int flags) {
  dim3 g((N + 127) / 128, (Mcap + 63) / 64);
  gemm_wmma<<<g, 256, 0, s>>>(A, lda, aShift, B, bOff, ldbK, ldbN, bias, res, ldr,
                              C, ldc, Mfix, Mdyn, N, Kfix, Kdyn, flags);
}

extern "C" void kernel_launch(void* const* d_in, const int* in_sizes, int n_in,
                              void* d_out, int out_size, void* d_ws, size_t ws_size,
                              hipStream_t stream) {
  const float* ff      = (const float*)d_in[0];
  const float* act     = (const float*)d_in[1];
  const float* fpos    = (const float*)d_in[2];
  const float* apos    = (const float*)d_in[3];
  const float* text    = (const float*)d_in[4];
  const float* Wf2c    = (const float*)d_in[5];
  const float* bf2c    = (const float*)d_in[6];
  const float* Wa2c    = (const float*)d_in[7];
  const float* ba2c    = (const float*)d_in[8];
  const float* Ws2c    = (const float*)d_in[9];
  const float* bs2c    = (const float*)d_in[10];
  const float* WihF    = (const float*)d_in[11];
  const float* WhhF    = (const float*)d_in[12];
  const float* bihF    = (const float*)d_in[13];
  const float* bhhF    = (const float*)d_in[14];
  const float* WihB    = (const float*)d_in[15];
  const float* WhhB    = (const float*)d_in[16];
  const float* bihB    = (const float*)d_in[17];
  const float* bhhB    = (const float*)d_in[18];
  const float* Wcomb   = (const float*)d_in[19];
  const float* bcomb   = (const float*)d_in[20];
  const float* f2aWq   = (const float*)d_in[21];
  const float* f2aWk   = (const float*)d_in[22];
  const float* f2aWv   = (const float*)d_in[23];
  const float* f2aWo   = (const float*)d_in[24];
  const float* f2abo   = (const float*)d_in[25];
  const float* agWih   = (const float*)d_in[26];
  const float* agWhh   = (const float*)d_in[27];
  const float* agbih   = (const float*)d_in[28];
  const float* agbhh   = (const float*)d_in[29];
  const float* a2fWq   = (const float*)d_in[30];
  const float* a2fWk   = (const float*)d_in[31];
  const float* a2fWv   = (const float*)d_in[32];
  const float* a2fWo   = (const float*)d_in[33];
  const float* a2fbo   = (const float*)d_in[34];
  const float* Wsf     = (const float*)d_in[35];
  const float* bsf     = (const float*)d_in[36];
  const float* WtcnIn  = (const float*)d_in[37];
  const float* btcnIn  = (const float*)d_in[38];
  const float* tcn_dW  = (const float*)d_in[39];
  const float* tcn_db  = (const float*)d_in[40];
  const float* tcn_pW  = (const float*)d_in[41];
  const float* tcn_pb  = (const float*)d_in[42];
  const float* WtcnOut = (const float*)d_in[43];
  const float* btcnOut = (const float*)d_in[44];

  float* outF = (float*)d_out;
  const size_t OFF_AF   = (size_t)T_SZ * 512;
  const size_t OFF_FSIM = OFF_AF + (size_t)64 * 512;
  const size_t OFF_ASIM = OFF_FSIM + (size_t)T_SZ * 32;
  const size_t OFF_SSIM = OFF_ASIM + (size_t)64 * 32;
  float* ffOut  = outF;
  float* afOut  = outF + OFF_AF;

  // -------- workspace carve --------
  char* w = (char*)d_ws;
  size_t cur = 0;
  auto allocF = [&](size_t n) { float* p = (float*)(w + cur); cur += n * sizeof(float); return p; };
  auto allocI = [&](size_t n) { int* p = (int*)(w + cur); cur += n * sizeof(int); return p; };
  float* tnorm   = allocF((size_t)C_CLS * 512);
  float* proj    = allocF((size_t)T_SZ * 512);
  float* seg     = allocF((size_t)SMAX * 512);
  float* segpos  = allocF((size_t)SMAX * 512);
  float* gi      = allocF((size_t)SMAX * 768);
  float* hf      = allocF((size_t)SMAX * 256);
  float* hb      = allocF((size_t)SMAX * 256);
  float* segcat  = allocF((size_t)SMAX * 512);
  float* segfeat = allocF((size_t)SMAX * 512);
  float* qbuf    = allocF((size_t)SMAX * 512);
  float* kbuf    = allocF((size_t)SMAX * 512);
  float* vbuf    = allocF((size_t)SMAX * 512);
  float* logits  = allocF((size_t)SMAX * 64);   // also T*32 sim scratch (equal size)
  float* seg2    = allocF((size_t)SMAX * 512);
  float* sm0     = allocF((size_t)64 * 512);
  float* sm1     = allocF((size_t)64 * 512);
  float* sm2     = allocF((size_t)64 * 512);
  float* agg     = allocF((size_t)64 * 1536);
  float* cat2    = allocF((size_t)T_SZ * 1024);
  float* xb      = allocF((size_t)T_SZ * 512);
  float* yb      = allocF((size_t)T_SZ * 512);
  int* pred    = allocI(T_SZ);
  int* segid   = allocI(T_SZ);
  int* starts  = allocI(SMAX);
  int* centers = allocI(SMAX);
  int* Sdev    = allocI(4);

  const float ATT_SCALE = 0.04419417382415922f;  // 1/sqrt(512)

  // ---- 1) segmentation: sim(ff) -> argmax -> runs ----
  norm_text<<<C_CLS, 256, 0, stream>>>(text, tnorm);
  gemm(stream, ff, 512, 0, Wf2c, 0, 512, 1, bf2c, nullptr, 0,
       proj, 512, T_SZ, T_SZ, nullptr, 512, 512, nullptr, 0);
  sim_rows<<<T_SZ, 256, 0, stream>>>(proj, tnorm, logits, T_SZ, nullptr, pred);
  segment_scan<<<1, 1024, 0, stream>>>(pred, segid, starts, centers, Sdev);
  seg_mean<<<SMAX, 256, 0, stream>>>(ff, starts, Sdev, seg);
  gather_pos<<<SMAX, 256, 0, stream>>>(fpos, centers, Sdev, segpos);

  // ---- 2) bidirectional segment GRU + combine ----
  gemm(stream, seg, 512, 0, WihF, 0, 768, 1, bihF, nullptr, 0,
       gi, 768, SMAX, 0, Sdev, 768, 512, nullptr, 0);
  gru_scan<<<1, 256, 256 * sizeof(float), stream>>>(gi, WhhF, bhhF, hf, 256, 0, Sdev, 0);
  gemm(stream, seg, 512, 0, WihB, 0, 768, 1, bihB, nullptr, 0,
       gi, 768, SMAX, 0, Sdev, 768, 512, nullptr, 0);
  gru_scan<<<1, 256, 256 * sizeof(float), stream>>>(gi, WhhB, bhhB, hb, 256, 0, Sdev, 1);
  segcat_relu<<<(SMAX * 512) / 256, 256, 0, stream>>>(hf, hb, segcat, Sdev);
  gemm(stream, segcat, 512, 0, Wcomb, 0, 512, 1, bcomb, nullptr, 0,
       segfeat, 512, SMAX, 0, Sdev, 512, 512, nullptr, 0);

  // ---- 3) f2a cross-attention (actions query segments) ----
  add_rows<<<(64 * 512 + 255) / 256, 256, 0, stream>>>(act, apos, sm0, 64, nullptr);
  gemm(stream, sm0, 512, 0, f2aWq, 0, 512, 1, nullptr, nullptr, 0,
       sm1, 512, 64, 64, nullptr, 512, 512, nullptr, 0);
  add_rows<<<(SMAX * 512) / 256, 256, 0, stream>>>(segfeat, segpos, qbuf, 0, Sdev);
  gemm(stream, qbuf, 512, 0, f2aWk, 0, 512, 1, nullptr, nullptr, 0,
       kbuf, 512, SMAX, 0, Sdev, 512, 512, nullptr, 0);
  gemm(stream, segfeat, 512, 0, f2aWv, 0, 512, 1, nullptr, nullptr, 0,
       vbuf, 512, SMAX, 0, Sdev, 512, 512, nullptr, 0);
  gemm(stream, sm1, 512, 0, kbuf, 0, 1, 512, nullptr, nullptr, 0,          // q @ k^T
       logits, SMAX, 64, 64, nullptr, SMAX, 512, nullptr, 0);
  softmax_rows<<<64, 256, 0, stream>>>(logits, SMAX, 64, nullptr, 0, Sdev, ATT_SCALE);
  gemm(stream, logits, SMAX, 0, vbuf, 0, 512, 1, nullptr, nullptr, 0,      // attn @ v
       sm2, 512, 64, 64, nullptr, 512, 0, Sdev, 0);
  gemm(stream, sm2, 512, 0, f2aWo, 0, 512, 1, f2abo, nullptr, 0,
       sm0, 512, 64, 64, nullptr, 512, 512, nullptr, 0);

  // ---- 4) action GRU (H=512, 64 steps) -> af output ----
  gemm(stream, sm0, 512, 0, agWih, 0, 1536, 1, agbih, nullptr, 0,
       agg, 1536, 64, 64, nullptr, 1536, 512, nullptr, 0);
  gru_scan<<<1, 512, 512 * sizeof(float), stream>>>(agg, agWhh, agbhh, afOut, 512, 64, nullptr, 0);

  // ---- 5) a2f cross-attention (segments query actions) ----
  gemm(stream, qbuf, 512, 0, a2fWq, 0, 512, 1, nullptr, nullptr, 0,        // (seg+pos) @ Wq
       vbuf, 512, SMAX, 0, Sdev, 512, 512, nullptr, 0);
  add_rows<<<(64 * 512 + 255) / 256, 256, 0, stream>>>(afOut, apos, sm0, 64, nullptr);
  gemm(stream, sm0, 512, 0, a2fWk, 0, 512, 1, nullptr, nullptr, 0,
       sm1, 512, 64, 64, nullptr, 512, 512, nullptr, 0);
  gemm(stream, afOut, 512, 0, a2fWv, 0, 512, 1, nullptr, nullptr, 0,
       sm2, 512, 64, 64, nullptr, 512, 512, nullptr, 0);
  gemm(stream, vbuf, 512, 0, sm1, 0, 1, 512, nullptr, nullptr, 0,          // q @ k^T (N=64)
       logits, 64, SMAX, 0, Sdev, 64, 512, nullptr, 0);
  softmax_rows<<<SMAX, 256, 0, stream>>>(logits, 64, 0, Sdev, 64, nullptr, ATT_SCALE);
  gemm(stream, logits, 64, 0, sm2, 0, 512, 1, nullptr, nullptr, 0,         // attn @ v
       qbuf, 512, SMAX, 0, Sdev, 512, 64, nullptr, 0);
  gemm(stream, qbuf, 512, 0, a2fWo, 0, 512, 1, a2fbo, nullptr, 0,
       seg2, 512, SMAX, 0, Sdev, 512, 512, nullptr, 0);

  // ---- 6) upsample + fuse, then TCN ----
  upsample_cat<<<(T_SZ * 512) / 256, 256, 0, stream>>>(seg2, ff, segid, cat2);
  gemm(stream, cat2, 1024, 0, Wsf, 0, 512, 1, bsf, nullptr, 0,
       proj, 512, T_SZ, T_SZ, nullptr, 512, 1024, nullptr, GF_RELU);
  gemm(stream, proj, 512, 0, WtcnIn, 0, 1, 512, btcnIn, nullptr, 0,        // 1x1 in (W^T)
       xb, 512, T_SZ, T_SZ, nullptr, 512, 512, nullptr, 0);
  for (int i = 0; i < L_TCN; ++i) {
    int dil = 1 << i;
    const float* dW = tcn_dW + (size_t)i * 512 * 512 * 3;
    const float* db = tcn_db + (size_t)i * 512;
    const float* pW = tcn_pW + (size_t)i * 512 * 512;
    const float* pb = tcn_pb + (size_t)i * 512;
    // y = relu(dilated_conv3(x) + db): three shifted GEMM taps, accumulate
    gemm(stream, xb, 512, -dil, dW, 0, 3, 1536, db, nullptr, 0,
         yb, 512, T_SZ, T_SZ, nullptr, 512, 512, nullptr, 0);
    gemm(stream, xb, 512, 0,    dW, 1, 3, 1536, nullptr, nullptr, 0,
         yb, 512, T_SZ, T_SZ, nullptr, 512, 512, nullptr, GF_ACC);
    gemm(stream, xb, 512, +dil, dW, 2, 3, 1536, nullptr, nullptr, 0,
         yb, 512, T_SZ, T_SZ, nullptr, 512, 512, nullptr, GF_ACC | GF_RELU);
    // x = x + 1x1(y) + pb   (in-place residual: read/write same element only)
    gemm(stream, yb, 512, 0, pW, 0, 1, 512, pb, xb, 512,
         xb, 512, T_SZ, T_SZ, nullptr, 512, 512, nullptr, 0);
  }
  gemm(stream, xb, 512, 0, WtcnOut, 0, 1, 512, btcnOut, nullptr, 0,        // 1x1 out -> ff_out
       ffOut, 512, T_SZ, T_SZ, nullptr, 512, 512, nullptr, 0);

  // ---- 7) output sims ----
  gemm(stream, ffOut, 512, 0, Wf2c, 0, 512, 1, bf2c, nullptr, 0,
       proj, 512, T_SZ, T_SZ, nullptr, 512, 512, nullptr, 0);
  sim_rows<<<T_SZ, 256, 0, stream>>>(proj, tnorm, outF + OFF_FSIM, T_SZ, nullptr, nullptr);
  gemm(stream, afOut, 512, 0, Wa2c, 0, 512, 1, ba2c, nullptr, 0,
       sm0, 512, 64, 64, nullptr, 512, 512, nullptr, 0);
  sim_rows<<<64, 256, 0, stream>>>(sm0, tnorm, outF + OFF_ASIM, 64, nullptr, nullptr);
  gemm(stream, seg2, 512, 0, Ws2c, 0, 512, 1, bs2c, nullptr, 0,
       vbuf, 512, SMAX, 0, Sdev, 512, 512, nullptr, 0);
  sim_rows<<<SMAX, 256, 0, stream>>>(vbuf, tnorm, outF + OFF_SSIM, 0, Sdev, nullptr);

  (void)in_sizes; (void)n_in; (void)out_size; (void)ws_size;
}